// diff_lpc2rc_86260123174591
// MI455X (gfx1250) — compile-verified
//
#include <hip/hip_runtime.h>

#define LPC_N 16
#define THREADS 256
#define ROWS_PER_BLOCK 256
#define ROW_PAD_F 18   // 72 B/row in LDS -> conflict-free ds_load_b64 reads

typedef float v4f __attribute__((ext_vector_type(4)));   // native vector: OK for nontemporal builtin
typedef float v2f __attribute__((ext_vector_type(2)));

__global__ __launch_bounds__(THREADS)
void lpc2rc_stepdown_kernel(const float* __restrict__ x,
                            float* __restrict__ out,
                            int nrows) {
  __shared__ __align__(16) float sm[ROWS_PER_BLOCK * ROW_PAD_F];

  const int t = threadIdx.x;
  const int block_row0 = blockIdx.x * ROWS_PER_BLOCK;

  int rows_here = nrows - block_row0;
  if (rows_here < 0) rows_here = 0;
  if (rows_here > ROWS_PER_BLOCK) rows_here = ROWS_PER_BLOCK;
  const int chunks_here = rows_here * 4;          // 16-byte chunks in this block

  const float* gbase = x + (long long)block_row0 * LPC_N;
  // Low 32 bits of a generic pointer to LDS == wave-relative LDS byte offset.
  const unsigned lds_base = (unsigned)(unsigned long long)(const void*)sm;

  // ---- Stage the block's input tile into LDS with async DMA-style copies.
  // Each wave instruction moves a dense, contiguous 512B (32 lanes x 16B).
#pragma unroll
  for (int c = 0; c < 4; ++c) {
    const int g = c * THREADS + t;                // chunk id within block (contiguous per instr)
    if (g < chunks_here) {
      const int row = g >> 2;
      const int ch  = g & 3;
      const unsigned lds_off =
          lds_base + (unsigned)(row * (ROW_PAD_F * 4) + ch * 16);
      const unsigned long long ga = (unsigned long long)(gbase + g * 4);
      asm volatile("global_load_async_to_lds_b128 %0, %1, off"
                   :: "v"(lds_off), "v"(ga)
                   : "memory");
    }
  }
  asm volatile("s_wait_asynccnt 0" ::: "memory");
  __syncthreads();

  if (t < rows_here) {
    // ---- Conflict-free row fetch from LDS (bank = 2*((9*lane + k) mod 32)).
    float r[LPC_N];
    const v2f* lrow = (const v2f*)(sm + t * ROW_PAD_F);
#pragma unroll
    for (int k = 0; k < LPC_N / 2; ++k) {
      v2f v = lrow[k];
      r[2 * k]     = v.x;
      r[2 * k + 1] = v.y;
    }

    // ---- Step-down recursion with deferred reciprocal scaling.
    // Active prefix r[0..LPC_N-i] carries common scale c; real value = c*r[k].
    float c = 1.0f;
#pragma unroll
    for (int i = 1; i < LPC_N; ++i) {
      const int m = LPC_N - i;            // prefix length to update; ki index = m
      const float ki = c * r[m];          // real reflection coefficient
      r[m] = ki;                          // element leaves active set: finalize in place
#pragma unroll
      for (int k = 0; k < m / 2; ++k) {
        const float lo = r[k];
        const float hi = r[m - 1 - k];
        r[k]         = fmaf(-ki, hi, lo); // unscaled numerator: a[k] - ki*a[rev]
        r[m - 1 - k] = fmaf(-ki, lo, hi);
      }
      if (m & 1) {
        const float mid = r[m / 2];
        r[m / 2] = fmaf(-ki, mid, mid);
      }
      const float den = fmaf(-ki, ki, 1.0f);          // 1 - ki^2
      float inv = __builtin_amdgcn_rcpf(den);         // v_rcp_f32
      inv = inv * fmaf(-den, inv, 2.0f);              // 1 Newton step -> ~0.5 ulp
      c *= inv;                                       // fold division into running scale
    }
    r[0] *= c;                                        // finalize last active element

    // ---- Streaming 128-bit non-temporal stores (output won't be re-read).
    float* orow = out + (long long)(block_row0 + t) * LPC_N;
#pragma unroll
    for (int k = 0; k < 4; ++k) {
      v4f v = { r[4 * k], r[4 * k + 1], r[4 * k + 2], r[4 * k + 3] };
      __builtin_nontemporal_store(v, (v4f*)orow + k);
    }
  }
}

extern "C" void kernel_launch(void* const* d_in, const int* in_sizes, int n_in,
                              void* d_out, int out_size, void* d_ws, size_t ws_size,
                              hipStream_t stream) {
  const float* x = (const float*)d_in[0];
  float* out = (float*)d_out;
  const long long nrows = (long long)in_sizes[0] / LPC_N;   // 256*8000 = 2,048,000
  const int grid = (int)((nrows + ROWS_PER_BLOCK - 1) / ROWS_PER_BLOCK);
  lpc2rc_stepdown_kernel<<<grid, THREADS, 0, stream>>>(x, out, (int)nrows);
}